// BERTInputRepresentation_69398081569261
// MI455X (gfx1250) — compile-verified
//
#include <hip/hip_runtime.h>

typedef float v2f __attribute__((ext_vector_type(2)));
typedef float v8f __attribute__((ext_vector_type(8)));

// Problem constants (fixed by the reference harness)
#define BB   64
#define SS   512
#define FF   64     // K dimension of the GEMM
#define DD   1024   // N dimension
#define LDSP 68     // padded LDS row stride (floats): conflict-free A-frag loads

// Fused: out[b,s,d] = x1 + seg + pe, where x1 = x@W + b_emb,
// seg[segment,d] = sum_j w_seg[j]*x1[row_j,d] + b_seg, pe = sinusoidal(s,d).
// One block = 16 rows x 128 cols; one wave = one 16x16 WMMA C tile.
__global__ __launch_bounds__(256) void bert_input_fused_kernel(
    const float* __restrict__ x,      // [B*S, F]
    const float* __restrict__ W,      // [F, D]
    const float* __restrict__ b_emb,  // [D]
    const float* __restrict__ w_seg,  // [8]
    const float* __restrict__ b_seg,  // [1]
    float* __restrict__ out)          // [B*S, D]
{
    __shared__ float xs[16 * LDSP];

    const int rowbase = blockIdx.x << 4;   // 16-row tile (never crosses a batch: 512%16==0)
    const int colgrp  = blockIdx.y << 7;   // 128-col group

    // ---- Stage the 16x64 x-tile into LDS (row-major, padded stride 68) ----
    {
        const int t  = threadIdx.x;
        const int r  = t >> 4;          // 0..15
        const int k4 = (t & 15) << 2;   // 0,4,...,60
        const float4 v = *(const float4*)(x + (rowbase + r) * FF + k4);
        *(float4*)(&xs[r * LDSP + k4]) = v;   // ds_store_b128, full 64-bank coverage
    }
    __syncthreads();

    const int wave = threadIdx.x >> 5;
    const int lane = threadIdx.x & 31;
    const int half = lane >> 4;        // 0: rows 0..7 / K even-pair, 1: rows 8..15
    const int m    = lane & 15;        // row (A) / column (B,C)
    const int col  = colgrp + (wave << 4) + m;

    // ---- GEMM: 16 chained v_wmma_f32_16x16x4_f32, K = 64 ----
    v8f c = {};
#pragma unroll
    for (int kk = 0; kk < 16; ++kk) {
        const int k = (kk << 2) + (half << 1);   // A/B fragment K base for this lane
        v2f a, b;
        // A 16x4 layout: lane m holds A[m][k], A[m][k+1] (half selects K pair)
        a.x = xs[m * LDSP + k];
        a.y = xs[m * LDSP + k + 1];
        // B 4x16 layout (mirrored): lane holds W[k][col], W[k+1][col]
        b.x = W[k * DD + col];
        b.y = W[(k + 1) * DD + col];
        c = __builtin_amdgcn_wmma_f32_16x16x4_f32(
                /*neg_a=*/false, a, /*neg_b=*/false, b,
                /*c_mod=*/(short)0, c, /*reuse_a=*/false, /*reuse_b=*/false);
    }

    // ---- Segment embedding: this lane's 8 accumulators ARE one segment ----
    // x1[j] = c[j] + b_emb[col];  seg = b_seg + sum_j w_seg[j]*x1[j]
    const float be = b_emb[col];
    float wloc[8];
    float wsum = 0.f;
#pragma unroll
    for (int j = 0; j < 8; ++j) { wloc[j] = w_seg[j]; wsum += wloc[j]; }
    float seg = b_seg[0] + be * wsum;
#pragma unroll
    for (int j = 0; j < 8; ++j) seg += wloc[j] * c[j];

    // ---- Sinusoidal PE (branchless): cos(x) = sin(x + pi/2) ----
    // div = exp(-ln(10000)/D * 2k), k = col/2 ; ln(10000) = 9.210340371976184
    const float dv    = __expf((float)(col >> 1) * (-2.0f * 9.210340371976184f / (float)DD));
    const float phase = (float)(col & 1) * 1.5707963267948966f;  // odd cols -> cos
    const int   s0    = (rowbase + (half << 3)) & (SS - 1);      // position within sequence

    float* outp = out + (size_t)(rowbase + (half << 3)) * DD + col;
#pragma unroll
    for (int j = 0; j < 8; ++j) {
        const float ang = __builtin_fmaf((float)(s0 + j), dv, phase);
        const float pe  = __sinf(ang);                 // single trans op, no divergence
        // Output is write-once streaming data: non-temporal store keeps L2
        // free for the reused x tile (8x) and W_emb (2048x).
        __builtin_nontemporal_store(c[j] + be + seg + pe, outp + (size_t)j * DD);
    }
}

extern "C" void kernel_launch(void* const* d_in, const int* in_sizes, int n_in,
                              void* d_out, int out_size, void* d_ws, size_t ws_size,
                              hipStream_t stream) {
    const float* x     = (const float*)d_in[0];
    const float* W_emb = (const float*)d_in[1];
    const float* b_emb = (const float*)d_in[2];
    const float* w_seg = (const float*)d_in[3];
    const float* b_seg = (const float*)d_in[4];
    float* out = (float*)d_out;

    // rows: B*S/16 = 2048 tiles; cols: D/128 = 8 groups
    dim3 grid(2048, 8);
    dim3 block(256);
    hipLaunchKernelGGL(bert_input_fused_kernel, grid, block, 0, stream,
                       x, W_emb, b_emb, w_seg, b_seg, out);
}